// Raindrop_59253368815812
// MI455X (gfx1250) — compile-verified
//
#include <hip/hip_runtime.h>
#include <hip/hip_bf16.h>

// ---------------------------------------------------------------------------
// Raindrop layer for MI455X (gfx1250, wave32).
//
// Node-level fold (see round 0): all GEMMs are per-node, done with
// V_WMMA_F32_16X16X4_F32.  B-operands staged through LDS by the Tensor Data
// Mover (tensor_load_to_lds, double-buffered, s_wait_tensorcnt), with TDM LDS
// padding (32+8 dwords/row) making the ds_load fragment reads bank-conflict-
// free for both half-waves.
// ---------------------------------------------------------------------------

typedef __attribute__((ext_vector_type(2))) float    v2f;
typedef __attribute__((ext_vector_type(8))) float    v8f;
typedef __attribute__((ext_vector_type(4))) unsigned u32x4;
typedef __attribute__((ext_vector_type(8))) int      i32x8;
typedef __attribute__((ext_vector_type(4))) int      i32x4;

#define NNODES 20000
#define INCH   128
#define OUTCH  128
#define NSTEP  32
#define NEDGES 256000
#define KSEL   (NEDGES / 2)

#define ROWSTRIDE 40            // 32 data dwords + 8 pad dwords (TDM pad)
#define NSLICES   36            // 4 lin_w slices + 32 inc_w slices

// monotonic float<->uint order-preserving encoding
__device__ __forceinline__ unsigned f2ord(float f) {
  unsigned u = __float_as_uint(f);
  return (u & 0x80000000u) ? ~u : (u | 0x80000000u);
}
__device__ __forceinline__ float ord2f(unsigned k) {
  unsigned u = (k & 0x80000000u) ? (k & 0x7fffffffu) : ~k;
  return __uint_as_float(u);
}

struct SelState { unsigned prefix; unsigned remaining; unsigned tie; unsigned cnt; };

// ---------------------------------------------------------------------------
// TDM issue: load a [128 rows x 32 cols] fp32 tile (row stride `stride0`
// elements) from global into LDS at byte offset lds_off, padding +8 dwords
// after every 32 data dwords (-> LDS row stride = 40 dwords).
// D# layout per CDNA5 ISA 8.3/8.4; groups 2/3 zero (2-D tensor, tile_dim2=0).
// 6-arg builtin form (clang-23 / therock-10.0 toolchain).
// ---------------------------------------------------------------------------
__device__ __forceinline__ void tdm_load_slice(const float* gptr, unsigned stride0,
                                               unsigned tdim0, unsigned lds_off)
{
  const unsigned long long ga = (unsigned long long)gptr;
  u32x4 g0;
  g0.x = 1u;                                   // count=1, is_restore=0, gather=0
  g0.y = lds_off;                              // lds_addr[31:0]
  g0.z = (unsigned)(ga & 0xFFFFFFFFull);       // global_addr[31:0]
  g0.w = (unsigned)((ga >> 32) & 0x1FFFFFFull) // global_addr[56:32]
       | (2u << 30);                           // type=2 ("image")
  i32x8 g1;
  // data_size=2 (4B) | pad_enable | pad_interval=4 (32 dw) | pad_amount=7 (8 dw)
  g1[0] = (int)((2u << 16) | (1u << 20) | (4u << 22) | (7u << 25));
  g1[1] = (int)((tdim0 & 0xFFFFu) << 16);                    // tensor_dim0 lo
  g1[2] = (int)((tdim0 >> 16) | (128u << 16));               // dim0 hi | tensor_dim1=128
  g1[3] = (int)(32u << 16);                                  // tile_dim0 = 32
  g1[4] = (int)128u;                                         // tile_dim1 = 128 (tile_dim2=0)
  g1[5] = (int)stride0;                                      // tensor_dim0_stride lo
  g1[6] = 0;                                                 // stride hi | dim1_stride lo
  g1[7] = 0;
  i32x4 gz4; gz4[0] = 0; gz4[1] = 0; gz4[2] = 0; gz4[3] = 0;
  i32x8 gz8;
#pragma unroll
  for (int i = 0; i < 8; ++i) gz8[i] = 0;
  __builtin_amdgcn_tensor_load_to_lds(g0, g1, gz4, gz4, gz8, 0);
}

// ---------------------------------------------------------------------------
// K1: per-node WMMA GEMMs; one wave32 per 16-node tile, 8 waves/block.
// All waves consume identical weight slices -> TDM double-buffer in LDS.
// ---------------------------------------------------------------------------
__global__ __launch_bounds__(256) void k1_node_gemm(
    const float* __restrict__ x,      const float* __restrict__ p_t,
    const float* __restrict__ lin_w,  const float* __restrict__ lin_b,
    const float* __restrict__ inc_w,  const float* __restrict__ inc_b,
    const float* __restrict__ mw,
    float* __restrict__ v_node, float* __restrict__ node_beta,
    float* __restrict__ nbm)
{
  __shared__ float sbuf[2][128 * ROWSTRIDE];   // 2 x 20 KB

  const int wave = threadIdx.x >> 5;
  const int tile = blockIdx.x * 8 + wave;
  const bool active = (tile * 16 < NNODES);    // wave-uniform
  const int lane = threadIdx.x & 31;
  const int half = lane >> 4;
  const int n    = lane & 15;

  // ---- A fragments: 16 rows x 128 K, v2f per lane per K-step ----
  v2f a[32];
  float mwreg[8], bsum[8];
  if (active) {
    const int rowA = tile * 16 + n;
#pragma unroll
    for (int ks = 0; ks < 32; ++ks) {
      const int c = ks * 4 + half * 2;
      a[ks].x = x[rowA * INCH + c];
      a[ks].y = x[rowA * INCH + c + 1];
    }
#pragma unroll
    for (int v = 0; v < 8; ++v) {
      mwreg[v] = mw[(tile * 16 + v + 8 * half) * 16 + n];
      bsum[v]  = 0.0f;
    }
  }

  // ---- TDM pipeline: slice s<4 -> lin_w cols [32s,32s+32); else inc_w t=s-4 ----
  if (wave == 0) {
    tdm_load_slice(lin_w, 128u, 128u, (unsigned)(unsigned long long)&sbuf[0][0]);
  }

  for (int s = 0; s < NSLICES; ++s) {
    if (wave == 0) {
      if (s + 1 < NSLICES) {
        const int s1 = s + 1;
        const float* g = (s1 < 4) ? (lin_w + s1 * 32) : (inc_w + (s1 - 4) * 32);
        const unsigned st = (s1 < 4) ? 128u : 1024u;
        tdm_load_slice(g, st, st, (unsigned)(unsigned long long)&sbuf[s1 & 1][0]);
        __builtin_amdgcn_s_wait_tensorcnt(1);  // oldest (slice s) complete
      } else {
        __builtin_amdgcn_s_wait_tensorcnt(0);
      }
    }
    __syncthreads();                           // buf[s&1] ready for everyone

    if (active) {
      const float* __restrict__ buf = &sbuf[s & 1][0];
      // bias init for the two 16-col chains of this slice
      float bias0, bias1;
      if (s < 4) { bias0 = lin_b[s * 32 + n];        bias1 = lin_b[s * 32 + 16 + n]; }
      else       { bias0 = inc_b[(s - 4) * 32 + n];  bias1 = inc_b[(s - 4) * 32 + 16 + n]; }
      v8f acc0, acc1;
#pragma unroll
      for (int v = 0; v < 8; ++v) { acc0[v] = bias0; acc1[v] = bias1; }

      // two independent WMMA chains; fragments prefetched 8 k-steps at a time
#pragma unroll
      for (int g = 0; g < 4; ++g) {
        v2f f0[8], f1[8];
#pragma unroll
        for (int j = 0; j < 8; ++j) {
          const int kr = (g * 8 + j) * 4 + half * 2;
          f0[j].x = buf[kr * ROWSTRIDE + n];
          f0[j].y = buf[(kr + 1) * ROWSTRIDE + n];
          f1[j].x = buf[kr * ROWSTRIDE + 16 + n];
          f1[j].y = buf[(kr + 1) * ROWSTRIDE + 16 + n];
        }
#pragma unroll
        for (int j = 0; j < 8; ++j) {
          const int ks = g * 8 + j;
          acc0 = __builtin_amdgcn_wmma_f32_16x16x4_f32(false, a[ks], false, f0[j],
                                                       (short)0, acc0, false, false);
          acc1 = __builtin_amdgcn_wmma_f32_16x16x4_f32(false, a[ks], false, f1[j],
                                                       (short)0, acc1, false, false);
        }
      }

      if (s < 4) {
        // v_node = relu(x @ lin_w + b); chains = n-blocks 2s, 2s+1
#pragma unroll
        for (int v = 0; v < 8; ++v) {
          const int m = tile * 16 + v + 8 * half;
          v_node[m * OUTCH + s * 32 + n]      = fmaxf(acc0[v], 0.0f);
          v_node[m * OUTCH + s * 32 + 16 + n] = fmaxf(acc1[v], 0.0f);
        }
      } else {
        // beta[t] = (hW . map_weights[node] + hPE . p_t[t]) / 32
        const int t = s - 4;
        const float pt = p_t[t * 16 + n];
#pragma unroll
        for (int v = 0; v < 8; ++v) {
          float r = acc0[v] * mwreg[v] + acc1[v] * pt;
#pragma unroll
          for (int off = 8; off >= 1; off >>= 1)
            r += __shfl_xor(r, off, 32);       // reduce over n within half-wave
          if (n == 0) {
            const float beta = r * (1.0f / 32.0f);
            node_beta[(tile * 16 + v + 8 * half) * NSTEP + t] = beta;
            bsum[v] += beta;
          }
        }
      }
    }
    __syncthreads();                           // all done reading buf[s&1]
  }

  if (active && n == 0) {
#pragma unroll
    for (int v = 0; v < 8; ++v)
      nbm[tile * 16 + v + 8 * half] = bsum[v] * (1.0f / 32.0f);
  }
}

// K2: all_edge_weights[e] = ew[e] * mean_t(node_beta[dst[e]]) -> ordered key
__global__ void k2_scores(const int* __restrict__ ei, const float* __restrict__ ew,
                          const float* __restrict__ nbm, unsigned* __restrict__ keys)
{
  const int e = blockIdx.x * blockDim.x + threadIdx.x;
  if (e >= NEDGES) return;
  keys[e] = f2ord(ew[e] * nbm[ei[NEDGES + e]]);
}

__global__ void k_init_state(SelState* st) {
  st->prefix = 0u; st->remaining = (unsigned)KSEL; st->tie = 0u; st->cnt = 0u;
}

// K3: radix-select (4 byte passes) for the exact K-th largest key
__global__ void k3_hist(const unsigned* __restrict__ keys, const SelState* st,
                        unsigned* __restrict__ hist, int pass)
{
  const int e = blockIdx.x * blockDim.x + threadIdx.x;
  if (e >= NEDGES) return;
  const unsigned key = keys[e];
  const int shift = 24 - 8 * pass;
  if (pass == 0 || (key >> (shift + 8)) == st->prefix)
    atomicAdd(&hist[(key >> shift) & 0xFFu], 1u);
}

__global__ void k3_scan(SelState* st, const unsigned* __restrict__ hist)
{
  unsigned rem = st->remaining, prefix = st->prefix;
  for (int b = 255; b >= 0; --b) {
    const unsigned c = hist[b];
    if (rem <= c || b == 0) { prefix = (prefix << 8) | (unsigned)b; break; }
    rem -= c;
  }
  st->prefix = prefix;
  st->remaining = rem;
}

// K4: compact selected edge ids (all > thr, plus `remaining` ties)
__global__ void k4_select(const unsigned* __restrict__ keys, SelState* st,
                          int* __restrict__ sel)
{
  const int e = blockIdx.x * blockDim.x + threadIdx.x;
  if (e >= NEDGES) return;
  const unsigned key = keys[e], thr = st->prefix;
  bool take = false;
  if (key > thr)       take = true;
  else if (key == thr) take = atomicAdd(&st->tie, 1u) < st->remaining;
  if (take) sel[atomicAdd(&st->cnt, 1u)] = e;
}

// K5: segment max of gamma over selected edges grouped by SRC node
__global__ void k5_max(const int* __restrict__ sel, const SelState* st,
                       const int* __restrict__ ei, const float* __restrict__ ew,
                       const float* __restrict__ node_beta, unsigned* __restrict__ mkey)
{
  const int i = blockIdx.x * blockDim.x + threadIdx.x;
  if (i >= (int)st->cnt) return;
  const int e = sel[i];
  const int s = ei[e], d = ei[NEDGES + e];
  const float w = ew[e];
  const float* nb = node_beta + d * NSTEP;
#pragma unroll 8
  for (int t = 0; t < NSTEP; ++t)
    atomicMax(&mkey[s * NSTEP + t], f2ord(nb[t] * w));
}

// K6: segment sum of exp(gamma - max)
__global__ void k6_den(const int* __restrict__ sel, const SelState* st,
                       const int* __restrict__ ei, const float* __restrict__ ew,
                       const float* __restrict__ node_beta,
                       const unsigned* __restrict__ mkey, float* __restrict__ den)
{
  const int i = blockIdx.x * blockDim.x + threadIdx.x;
  if (i >= (int)st->cnt) return;
  const int e = sel[i];
  const int s = ei[e], d = ei[NEDGES + e];
  const float w = ew[e];
  const float* nb = node_beta + d * NSTEP;
#pragma unroll 8
  for (int t = 0; t < NSTEP; ++t) {
    const float g = nb[t] * w;
    const float m = ord2f(mkey[s * NSTEP + t]);
    atomicAdd(&den[s * NSTEP + t], __expf(g - m));
  }
}

// K7: out[src] += relu_v[dst] * softmax(gamma)  (gamma repeats over 4 cols)
__global__ void k7_out(const int* __restrict__ sel, const SelState* st,
                       const int* __restrict__ ei, const float* __restrict__ ew,
                       const float* __restrict__ node_beta,
                       const unsigned* __restrict__ mkey, const float* __restrict__ den,
                       const float* __restrict__ v_node, float* __restrict__ out)
{
  const int i = blockIdx.x * blockDim.x + threadIdx.x;
  if (i >= (int)st->cnt) return;
  const int e = sel[i];
  const int s = ei[e], d = ei[NEDGES + e];
  const float w = ew[e];
  const float* nb = node_beta + d * NSTEP;
  const float* vr = v_node + d * OUTCH;
  float* orow = out + s * OUTCH;
#pragma unroll 4
  for (int t = 0; t < NSTEP; ++t) {
    const float g = nb[t] * w;
    const float m = ord2f(mkey[s * NSTEP + t]);
    const float coef = __expf(g - m) / den[s * NSTEP + t];
    const float4 v4 = *(const float4*)&vr[t * 4];
    atomicAdd(&orow[t * 4 + 0], v4.x * coef);
    atomicAdd(&orow[t * 4 + 1], v4.y * coef);
    atomicAdd(&orow[t * 4 + 2], v4.z * coef);
    atomicAdd(&orow[t * 4 + 3], v4.w * coef);
  }
}

// ---------------------------------------------------------------------------
extern "C" void kernel_launch(void* const* d_in, const int* in_sizes, int n_in,
                              void* d_out, int out_size, void* d_ws, size_t ws_size,
                              hipStream_t stream)
{
  (void)in_sizes; (void)n_in; (void)out_size; (void)ws_size;
  const float* x     = (const float*)d_in[0];
  const float* p_t   = (const float*)d_in[1];
  const int*   ei    = (const int*)  d_in[2];
  const float* ew    = (const float*)d_in[3];
  const float* lin_w = (const float*)d_in[4];
  const float* lin_b = (const float*)d_in[5];
  const float* inc_w = (const float*)d_in[6];
  const float* inc_b = (const float*)d_in[7];
  const float* mw    = (const float*)d_in[8];
  float* out = (float*)d_out;

  char* ws = (char*)d_ws;
  auto alloc = [&](size_t bytes) -> char* {
    char* p = ws; ws += (bytes + 255) & ~size_t(255); return p;
  };
  float*    v_node    = (float*)   alloc(sizeof(float) * NNODES * OUTCH);
  float*    node_beta = (float*)   alloc(sizeof(float) * NNODES * NSTEP);
  float*    nbm       = (float*)   alloc(sizeof(float) * NNODES);
  unsigned* keys      = (unsigned*)alloc(sizeof(unsigned) * NEDGES);
  unsigned* hist      = (unsigned*)alloc(sizeof(unsigned) * 256);
  SelState* st        = (SelState*)alloc(sizeof(SelState));
  int*      sel       = (int*)     alloc(sizeof(int) * KSEL);
  unsigned* mkey      = (unsigned*)alloc(sizeof(unsigned) * NNODES * NSTEP);
  float*    den       = (float*)   alloc(sizeof(float) * NNODES * NSTEP);

  (void)hipMemsetAsync(out,  0, sizeof(float) * NNODES * OUTCH, stream);
  (void)hipMemsetAsync(mkey, 0, sizeof(unsigned) * NNODES * NSTEP, stream);
  (void)hipMemsetAsync(den,  0, sizeof(float) * NNODES * NSTEP, stream);

  k1_node_gemm<<<(NNODES / 16 + 7) / 8, 256, 0, stream>>>(
      x, p_t, lin_w, lin_b, inc_w, inc_b, mw, v_node, node_beta, nbm);

  k2_scores<<<(NEDGES + 255) / 256, 256, 0, stream>>>(ei, ew, nbm, keys);

  k_init_state<<<1, 1, 0, stream>>>(st);
  for (int p = 0; p < 4; ++p) {
    (void)hipMemsetAsync(hist, 0, 256 * sizeof(unsigned), stream);
    k3_hist<<<(NEDGES + 255) / 256, 256, 0, stream>>>(keys, st, hist, p);
    k3_scan<<<1, 1, 0, stream>>>(st, hist);
  }
  k4_select<<<(NEDGES + 255) / 256, 256, 0, stream>>>(keys, st, sel);

  k5_max<<<(KSEL + 255) / 256, 256, 0, stream>>>(sel, st, ei, ew, node_beta, mkey);
  k6_den<<<(KSEL + 255) / 256, 256, 0, stream>>>(sel, st, ei, ew, node_beta, mkey, den);
  k7_out<<<(KSEL + 255) / 256, 256, 0, stream>>>(sel, st, ei, ew, node_beta, mkey, den,
                                                 v_node, out);
}